// EEGGNN_23639499997343
// MI455X (gfx1250) — compile-verified
//
#include <hip/hip_runtime.h>
#include <hip/hip_bf16.h>

#define NEG_SLOPE 0.01f
#define NUM_GRAPHS 4096

typedef __attribute__((ext_vector_type(2))) float v2f;
typedef __attribute__((ext_vector_type(8))) float v8f;

__device__ __forceinline__ float lrelu(float v) {
    return (v > 0.0f) ? v : v * NEG_SLOPE;
}

// Hardware f32 atomic add at device scope (cross-WGP coherent, executes in GL2).
// Inline asm guarantees global_atomic_add_f32 (no CAS-loop fallback).
__device__ __forceinline__ void atomic_add_f32(float* p, float v) {
    asm volatile("global_atomic_add_f32 %0, %1, off scope:SCOPE_DEV"
                 :: "v"(p), "v"(v) : "memory");
}

// ---------------------------------------------------------------------------
// fp32 WMMA GEMM, K/NOUT compile-time:  Y[M,NOUT] = post( pre(X)[M,K] @ W )
//   pre  (INBIAS=1):  a = lrelu(x + bin[k])      (fuses previous layer epilogue)
//   post (OUTBIAS=1): y = c + bout[n];  OUTACT=1: y = lrelu(y)
// M must be a multiple of 16.  One wave32 per 16x16 tile, K-loop fully
// unrolled, branchless clamped loads -> back-to-back v_wmma_f32_16x16x4_f32.
// ---------------------------------------------------------------------------
template <int K, int NOUT, int INBIAS, int OUTBIAS, int OUTACT>
__global__ __launch_bounds__(256)
void wmma_gemm(const float* __restrict__ X,
               const float* __restrict__ bin,
               const float* __restrict__ W,
               const float* __restrict__ bout,
               float* __restrict__ Y,
               int M)
{
    constexpr int KP = (K + 3) & ~3;          // padded K (mult of 4)
    constexpr int NT = (NOUT + 15) / 16;      // column tiles
    static_assert((K & 1) == 0 && K >= 2, "K must be even");

    const int wavesPerBlock = blockDim.x >> 5;
    const int waveId = blockIdx.x * wavesPerBlock + (threadIdx.x >> 5);
    const int lane = threadIdx.x & 31;
    const int nRowTiles = M >> 4;
    if (waveId >= nRowTiles * NT) return;     // wave-uniform

    const int rowTile = waveId / NT;
    const int colTile = waveId % NT;
    const int laneLo = lane & 15;
    const int laneHi = lane >> 4;

    const int aRow = (rowTile << 4) + laneLo;           // always < M
    const int bColRaw = (colTile << 4) + laneLo;
    const int bCol = (bColRaw < NOUT) ? bColRaw : (NOUT - 1);   // clamp
    const bool colOk = bColRaw < NOUT;

    const float* __restrict__ xrow = X + (long)aRow * K;

    v8f c = {};
    #pragma unroll
    for (int k0 = 0; k0 < KP; k0 += 4) {
        const int ka = k0 + 2 * laneHi;                  // even
        const int kc = (ka + 1 < K) ? ka : (K - 2);      // clamp

        // A: aligned float2 from this row; zero lanes with k >= K
        float2 av = *(const float2*)(xrow + kc);
        float ax = av.x, ay = av.y;
        if constexpr (INBIAS) {
            float2 bb = *(const float2*)(bin + kc);
            ax = lrelu(ax + bb.x);
            ay = lrelu(ay + bb.y);
        }
        v2f a, b;
        a.x = (ka     < K) ? ax : 0.0f;
        a.y = (ka + 1 < K) ? ay : 0.0f;

        // B: clamped row/col loads; no masking needed (A is zero where k>=K,
        // and out-of-range columns are simply never stored)
        b.x = W[(long)kc * NOUT + bCol];
        b.y = W[(long)(kc + 1) * NOUT + bCol];

        c = __builtin_amdgcn_wmma_f32_16x16x4_f32(false, a, false, b,
                                                  (short)0, c, false, false);
    }

    float bv;
    if constexpr (OUTBIAS) bv = colOk ? bout[bCol] : 0.0f;
    #pragma unroll
    for (int r = 0; r < 8; ++r) {
        const int mrow = (rowTile << 4) + r + 8 * laneHi;
        if (colOk) {
            float v = c[r];
            if constexpr (OUTBIAS) v += bv;
            if constexpr (OUTACT) v = lrelu(v);
            Y[(long)mrow * NOUT + bColRaw] = v;
        }
    }
}

// ---------------------------------------------------------------------------
__global__ void zero4_kernel(float4* __restrict__ A, long total4)
{
    long i = (long)blockIdx.x * blockDim.x + threadIdx.x;
    if (i < total4) A[i] = make_float4(0.f, 0.f, 0.f, 0.f);
}

// ---------------------------------------------------------------------------
// Edge gather/scale/scatter: agg[dst] += h[src] * ew, chunked so each wave
// touches few random rows and moves whole cachelines.  FOUT%4==0 -> float4
// chunks; otherwise (50) float2 chunks (rows only 8B-aligned).
// ---------------------------------------------------------------------------
template <int FOUT>
__global__ __launch_bounds__(256)
void edge_scatter(const float* __restrict__ H,
                  const int* __restrict__ src,
                  const int* __restrict__ dst,
                  const float* __restrict__ ew,
                  float* __restrict__ AGG,
                  int E)
{
    if constexpr ((FOUT & 3) == 0) {
        constexpr int CPE = FOUT / 4;
        long idx = (long)blockIdx.x * blockDim.x + threadIdx.x;
        if (idx >= (long)E * CPE) return;
        const int e = (int)(idx / CPE);
        const int f = (int)(idx % CPE) * 4;
        const int s = src[e], d = dst[e];
        const float w = ew[e];
        float4 hv = *(const float4*)(H + (long)s * FOUT + f);
        float* ad = AGG + (long)d * FOUT + f;
        atomic_add_f32(ad + 0, hv.x * w);
        atomic_add_f32(ad + 1, hv.y * w);
        atomic_add_f32(ad + 2, hv.z * w);
        atomic_add_f32(ad + 3, hv.w * w);
    } else {
        constexpr int CPE = FOUT / 2;                 // FOUT even (50)
        long idx = (long)blockIdx.x * blockDim.x + threadIdx.x;
        if (idx >= (long)E * CPE) return;
        const int e = (int)(idx / CPE);
        const int f = (int)(idx % CPE) * 2;
        const int s = src[e], d = dst[e];
        const float w = ew[e];
        float2 hv = *(const float2*)(H + (long)s * FOUT + f);
        float* ad = AGG + (long)d * FOUT + f;
        atomic_add_f32(ad + 0, hv.x * w);
        atomic_add_f32(ad + 1, hv.y * w);
    }
}

// ---------------------------------------------------------------------------
// Global add pool, fusing layer-4 epilogue: conv_out = lrelu(agg + b4),
// then reference applies lrelu once more before pooling.
// pooled[batch[n], f] += lrelu(lrelu(agg[n,f] + b4[f]))
// ---------------------------------------------------------------------------
__global__ __launch_bounds__(256)
void pool_kernel(const float* __restrict__ agg,
                 const float* __restrict__ b4,
                 const int* __restrict__ batch,
                 float* __restrict__ pooled,
                 int N, int F)
{
    long i = (long)blockIdx.x * blockDim.x + threadIdx.x;
    if (i >= (long)N * F) return;
    const int n = (int)(i / F);
    const int f = (int)(i % F);
    float v = lrelu(lrelu(agg[i] + b4[f]));
    atomic_add_f32(&pooled[(long)batch[n] * F + f], v);
}

// ---------------------------------------------------------------------------

static inline int idiv(long a, int b) { return (int)((a + b - 1) / b); }

extern "C" void kernel_launch(void* const* d_in, const int* in_sizes, int n_in,
                              void* d_out, int out_size, void* d_ws, size_t ws_size,
                              hipStream_t stream)
{
    const float* x     = (const float*)d_in[0];
    const int*   ei    = (const int*)  d_in[1];
    const float* ew    = (const float*)d_in[2];
    const int*   batch = (const int*)  d_in[3];
    const float* W1 = (const float*)d_in[4];  const float* b1 = (const float*)d_in[5];
    const float* W2 = (const float*)d_in[6];  const float* b2 = (const float*)d_in[7];
    const float* W3 = (const float*)d_in[8];  const float* b3 = (const float*)d_in[9];
    const float* W4 = (const float*)d_in[10]; const float* b4 = (const float*)d_in[11];
    const float* Wf1 = (const float*)d_in[12]; const float* bf1 = (const float*)d_in[13];
    const float* Wf2 = (const float*)d_in[14]; const float* bf2 = (const float*)d_in[15];
    const float* Wf3 = (const float*)d_in[16]; const float* bf3 = (const float*)d_in[17];

    const int N = in_sizes[0] / 6;        // 253952 nodes (multiple of 16)
    const int E = in_sizes[2];            // 4,000,000 edges
    const int* src = ei;                  // edge_index[0, :]
    const int* dsti = ei + E;             // edge_index[1, :]

    // Workspace carve-up (256B aligned)
    char* ws = (char*)d_ws;
    auto carve = [&](size_t bytes) {
        char* p = ws;
        ws += (bytes + 255) & ~(size_t)255;
        return (float*)p;
    };
    float* H    = carve((size_t)N * 64 * sizeof(float));   // x @ W
    float* bufA = carve((size_t)N * 64 * sizeof(float));   // agg ping
    float* bufB = carve((size_t)N * 64 * sizeof(float));   // agg pong
    float* pooled = carve((size_t)NUM_GRAPHS * 50 * sizeof(float));
    float* y1     = carve((size_t)NUM_GRAPHS * 30 * sizeof(float));
    float* y2     = carve((size_t)NUM_GRAPHS * 20 * sizeof(float));

    const int BT = 256;                   // 8 waves/block
    const int WPB = BT / 32;

    auto zero = [&](float* p, long elems) {
        zero4_kernel<<<idiv(elems / 4, BT), BT, 0, stream>>>((float4*)p, elems / 4);
    };
    auto ggrid = [&](int M, int NOUT) {
        return idiv((long)(M / 16) * ((NOUT + 15) / 16), WPB);
    };

    // ---- GCN layer 1: 6 -> 16   (h1 = x @ W1) ----
    wmma_gemm<6, 16, 0, 0, 0><<<ggrid(N, 16), BT, 0, stream>>>(x, nullptr, W1, nullptr, H, N);
    zero(bufA, (long)N * 16);
    edge_scatter<16><<<idiv((long)E * 4, BT), BT, 0, stream>>>(H, src, dsti, ew, bufA, E);

    // ---- GCN layer 2: 16 -> 32  (in = lrelu(agg1 + b1), fused) ----
    wmma_gemm<16, 32, 1, 0, 0><<<ggrid(N, 32), BT, 0, stream>>>(bufA, b1, W2, nullptr, H, N);
    zero(bufB, (long)N * 32);
    edge_scatter<32><<<idiv((long)E * 8, BT), BT, 0, stream>>>(H, src, dsti, ew, bufB, E);

    // ---- GCN layer 3: 32 -> 64 ----
    wmma_gemm<32, 64, 1, 0, 0><<<ggrid(N, 64), BT, 0, stream>>>(bufB, b2, W3, nullptr, H, N);
    zero(bufA, (long)N * 64);
    edge_scatter<64><<<idiv((long)E * 16, BT), BT, 0, stream>>>(H, src, dsti, ew, bufA, E);

    // ---- GCN layer 4: 64 -> 50 ----
    wmma_gemm<64, 50, 1, 0, 0><<<ggrid(N, 50), BT, 0, stream>>>(bufA, b3, W4, nullptr, H, N);
    zero(bufB, (long)N * 50);
    edge_scatter<50><<<idiv((long)E * 25, BT), BT, 0, stream>>>(H, src, dsti, ew, bufB, E);

    // ---- global add pool (fuses b4 + double leaky_relu) ----
    zero(pooled, (long)NUM_GRAPHS * 50);
    pool_kernel<<<idiv((long)N * 50, BT), BT, 0, stream>>>(bufB, b4, batch, pooled, N, 50);

    // ---- MLP head: 50 -> 30 -> 20 -> 2 (bias + lrelu fused) ----
    wmma_gemm<50, 30, 0, 1, 1><<<ggrid(NUM_GRAPHS, 30), BT, 0, stream>>>(pooled, nullptr, Wf1, bf1, y1, NUM_GRAPHS);
    wmma_gemm<30, 20, 0, 1, 1><<<ggrid(NUM_GRAPHS, 20), BT, 0, stream>>>(y1, nullptr, Wf2, bf2, y2, NUM_GRAPHS);
    wmma_gemm<20,  2, 0, 1, 1><<<ggrid(NUM_GRAPHS, 2), BT, 0, stream>>>(y2, nullptr, Wf3, bf3, (float*)d_out, NUM_GRAPHS);
}